// KineticStateEstimator_88089779241049
// MI455X (gfx1250) — compile-verified
//
#include <hip/hip_runtime.h>

// N = 8388608 independent 2x2 Kalman updates. Pure HBM-bandwidth-bound stream
// (~33 B + ~10 FLOPs per item). Strategy: 4 items/thread, all-128-bit
// non-temporal global accesses, and CDNA5 async-to-LDS staging for the P
// array (ASYNCcnt path) so P's in-flight data costs no VGPRs.

typedef float f4 __attribute__((ext_vector_type(4)));

#define TPB 256

__global__ void __launch_bounds__(TPB) kinetic_state_estimator_kernel(
    const float* __restrict__ x,          // (N,2,1)
    const float* __restrict__ P,          // (N,2,2)
    const float* __restrict__ z,          // (N,1)
    const unsigned char* __restrict__ mask, // (N,) bool
    float* __restrict__ out,              // (N,)
    int n)
{
    __shared__ float shP[TPB * 16];       // 16 KB: 4 items * 4 floats per thread
    const int tid = threadIdx.x;
    const long long b = ((long long)blockIdx.x * TPB + tid) * 4;

    if (b + 4 <= (long long)n) {
        // ---- CDNA5 async copy: stage this thread's 4 P matrices (64 B) into
        // LDS. inst_offset applies to both the LDS and global addresses, so a
        // single VGPR address pair drives all four B128 transfers. Tracked by
        // ASYNCcnt, overlapping the NT vector loads below.
        unsigned lds_off = (unsigned)(size_t)(&shP[tid * 16]);
        const float* pg = P + b * 4;
        asm volatile(
            "global_load_async_to_lds_b128 %0, %1, off\n\t"
            "global_load_async_to_lds_b128 %0, %1, off offset:16\n\t"
            "global_load_async_to_lds_b128 %0, %1, off offset:32\n\t"
            "global_load_async_to_lds_b128 %0, %1, off offset:48"
            :: "v"(lds_off), "v"(pg)
            : "memory");

        // ---- Streaming (non-temporal) 128-bit loads: data is touched once,
        // don't pollute the 192 MB L2.
        f4 xa = __builtin_nontemporal_load((const f4*)(x + 2 * b));      // x[b],x[b+1]
        f4 xb = __builtin_nontemporal_load((const f4*)(x + 2 * b + 4));  // x[b+2],x[b+3]
        f4 zv = __builtin_nontemporal_load((const f4*)(z + b));
        unsigned mv = __builtin_nontemporal_load((const unsigned*)(mask + b));

        asm volatile("s_wait_asynccnt 0" ::: "memory");

        const float* sp = &shP[tid * 16];
        f4 r;
#pragma unroll
        for (int k = 0; k < 4; ++k) {
            float x0 = (k == 0) ? xa[0] : (k == 1) ? xa[2] : (k == 2) ? xb[0] : xb[2];
            float x1 = (k == 0) ? xa[1] : (k == 1) ? xa[3] : (k == 2) ? xb[1] : xb[3];
            float p00 = sp[4 * k + 0];
            float p01 = sp[4 * k + 1];
            float p10 = sp[4 * k + 2];
            float p11 = sp[4 * k + 3];

            // F = [[1,1],[0,1]]: x_pred0 = x0+x1
            float xp0  = x0 + x1;
            // P_pred00 = p00+p01+p10+p11 + Q00
            float pp00 = (p00 + p01) + (p10 + p11) + 1.0e-4f;
            float y    = zv[k] - xp0;
            float S    = pp00 + 0.010001f;          // MEASURE_NOISE + EPS
            float K0   = pp00 / S;
            float xn   = fmaf(K0, y, xp0);
            bool  m    = ((mv >> (8 * k)) & 0xFFu) != 0u;
            r[k] = m ? xn : xp0;
        }
        __builtin_nontemporal_store(r, (f4*)(out + b));
    } else {
        // scalar tail (not hit for N = 8388608, kept for safety)
        for (long long i = b; i < (long long)n && i < b + 4; ++i) {
            float x0 = x[2 * i], x1 = x[2 * i + 1];
            float p00 = P[4 * i + 0], p01 = P[4 * i + 1];
            float p10 = P[4 * i + 2], p11 = P[4 * i + 3];
            float xp0  = x0 + x1;
            float pp00 = (p00 + p01) + (p10 + p11) + 1.0e-4f;
            float y    = z[i] - xp0;
            float S    = pp00 + 0.010001f;
            float K0   = pp00 / S;
            float xn   = fmaf(K0, y, xp0);
            out[i] = mask[i] ? xn : xp0;
        }
    }
}

extern "C" void kernel_launch(void* const* d_in, const int* in_sizes, int n_in,
                              void* d_out, int out_size, void* d_ws, size_t ws_size,
                              hipStream_t stream) {
    const float* x            = (const float*)d_in[0];
    const float* P            = (const float*)d_in[1];
    const float* z            = (const float*)d_in[2];
    const unsigned char* mask = (const unsigned char*)d_in[3];
    float* out                = (float*)d_out;

    const int n = in_sizes[2];                 // N (z has one element per item)
    const int nthreads = (n + 3) / 4;          // 4 items per thread
    const int blocks = (nthreads + TPB - 1) / TPB;

    kinetic_state_estimator_kernel<<<blocks, TPB, 0, stream>>>(x, P, z, mask, out, n);
}